// DeployModel_3444563771796
// MI455X (gfx1250) — compile-verified
//
#include <hip/hip_runtime.h>
#include <cstdint>

#define B_ 16
#define Q_ 20000
#define C_ 80
#define K_ 100

typedef __attribute__((ext_vector_type(4))) unsigned int uint32x4;
typedef __attribute__((ext_vector_type(8))) int          int32x8;
typedef __attribute__((ext_vector_type(4))) int          int32x4;

// ---------------------------------------------------------------------------
// Kernel 1: per-query max over the 80 class logits.
// sigmoid is strictly monotonic, so ranking by raw max-logit is identical to
// ranking by max(sigmoid(logit)).  Streams 102.4 MB -> bandwidth bound.
// ---------------------------------------------------------------------------
__global__ __launch_bounds__(256) void score_kernel(const float* __restrict__ logits,
                                                    float* __restrict__ scores,
                                                    int total) {
    int gid = blockIdx.x * blockDim.x + threadIdx.x;
    if (gid >= total) return;
    const float4* p = (const float4*)(logits + (size_t)gid * C_);
    float m = -INFINITY;
#pragma unroll
    for (int i = 0; i < C_ / 4; ++i) {
        float4 v = p[i];
        m = fmaxf(m, fmaxf(fmaxf(v.x, v.y), fmaxf(v.z, v.w)));
    }
    scores[gid] = m;
}

// better(v2,i2) beats (v1,i1): higher value, ties broken by LOWER index
// (matches jax.lax.top_k tie-breaking).
__device__ __forceinline__ bool better(float v2, int i2, float v1, int i1) {
    return (v2 > v1) || (v2 == v1 && i2 < i1);
}

// ---------------------------------------------------------------------------
// Kernel 2: one block per batch.  Stage 20000 scores into LDS (TDM descriptor
// load + per-lane async global->LDS copies), then iterative top-100 selection
// with cached per-thread local maxima, then gather boxes + sigmoid(logits).
// ---------------------------------------------------------------------------
__global__ __launch_bounds__(1024) void topk_kernel(const float* __restrict__ scores,
                                                    const float* __restrict__ logits,
                                                    const float* __restrict__ boxes,
                                                    float* __restrict__ out_boxes,
                                                    float* __restrict__ out_scores) {
    extern __shared__ float smem[];       // Q_ floats (80 KB dynamic LDS)
    __shared__ float cval[32];
    __shared__ int   cidx[32];
    __shared__ int   topIdx[K_];
    __shared__ int   bcast;

    const int b    = blockIdx.x;
    const int tid  = threadIdx.x;
    const int lane = tid & 31;
    const int wave = tid >> 5;

    const float*   gsrc     = scores + (size_t)b * Q_;
    const uint32_t lds_base = (uint32_t)(uintptr_t)(&smem[0]);

    // ---- Phase A: TDM bulk tile load (Tensor Data Mover, TENSORcnt path) ----
    // 1-row tile of 20000 f32 elements: tensor_dim0 = tile_dim0 = 20000,
    // data_size = 4 bytes, not in a cluster (workgroup_mask = 0).
#if __has_builtin(__builtin_amdgcn_tensor_load_to_lds)
    if (tid == 0) {
        uint64_t ga = (uint64_t)(uintptr_t)gsrc;
        uint32x4 g0;
        g0.x = 1u;                                        // count=1 (valid), user mode
        g0.y = lds_base;                                  // lds_addr[31:0]
        g0.z = (uint32_t)(ga & 0xffffffffu);              // global_addr[31:0]
        g0.w = (uint32_t)((ga >> 32) & 0x01ffffffu)       // global_addr[56:32]
             | (2u << 30);                                // type = 2 ("image")
        int32x8 g1;
        g1[0] = 0x00020000;                               // data_size=2 -> 4 bytes
        g1[1] = (int)(((uint32_t)Q_ & 0xffffu) << 16);    // tensor_dim0[15:0]
        g1[2] = (int)(((uint32_t)Q_ >> 16) & 0xffffu)     // tensor_dim0[31:16]
              | (1 << 16);                                // tensor_dim1 = 1
        g1[3] = (int)(((uint32_t)Q_ & 0xffffu) << 16);    // tile_dim0 = 20000
        g1[4] = 1;                                        // tile_dim1 = 1
        g1[5] = Q_;                                       // tensor_dim0_stride
        g1[6] = 0;
        g1[7] = 0;
        int32x4 gz = {0, 0, 0, 0};
#if defined(__clang_major__) && (__clang_major__ >= 23)
        int32x8 gz8 = {0, 0, 0, 0, 0, 0, 0, 0};
        __builtin_amdgcn_tensor_load_to_lds(g0, g1, gz, gz, gz8, 0);
#else
        __builtin_amdgcn_tensor_load_to_lds(g0, g1, gz, gz, 0);
#endif
#if __has_builtin(__builtin_amdgcn_s_wait_tensorcnt)
        __builtin_amdgcn_s_wait_tensorcnt(0);
#else
        asm volatile("s_wait_tensorcnt 0" ::: "memory");
#endif
    }
    __syncthreads();
#endif

    // ---- Phase B: per-lane async global->LDS copy (ASYNCcnt path) ----------
    // Deterministically (re)writes the same region, so correctness never
    // depends on the TDM descriptor above.
    for (int m = tid; m < Q_ / 4; m += 1024) {            // 5000 float4 tiles
        uint64_t ga = (uint64_t)(uintptr_t)(gsrc + m * 4);
        uint32_t la = lds_base + (uint32_t)m * 16u;
        asm volatile("global_load_async_to_lds_b128 %0, %1, off"
                     :: "v"(la), "v"(ga) : "memory");
    }
    asm volatile("s_wait_asynccnt 0" ::: "memory");
    __syncthreads();

    // ---- Initial per-thread local max over strided elements ----
    float v = -INFINITY;
    int   idx = 0x7fffffff;
    for (int q = tid; q < Q_; q += 1024) {
        float s = smem[q];
        if (better(s, q, v, idx)) { v = s; idx = q; }
    }

    // ---- Iterative selection of top-K (descending, low-index tie-break) ----
    for (int k = 0; k < K_; ++k) {
        // wave32 shuffle reduction of cached candidates
        float rv = v; int ri = idx;
#pragma unroll
        for (int off = 16; off; off >>= 1) {
            float ov = __shfl_xor(rv, off, 32);
            int   oi = __shfl_xor(ri, off, 32);
            if (better(ov, oi, rv, ri)) { rv = ov; ri = oi; }
        }
        if (lane == 0) { cval[wave] = rv; cidx[wave] = ri; }
        __syncthreads();

        // wave 0 reduces the 32 per-wave winners
        if (wave == 0) {
            float fv = cval[lane];
            int   fi = cidx[lane];
#pragma unroll
            for (int off = 16; off; off >>= 1) {
                float ov = __shfl_xor(fv, off, 32);
                int   oi = __shfl_xor(fi, off, 32);
                if (better(ov, oi, fv, fi)) { fv = ov; fi = oi; }
            }
            if (lane == 0) {
                topIdx[k] = fi;
                bcast     = fi;
                smem[fi]  = -INFINITY;   // remove winner
            }
        }
        __syncthreads();

        // only the owner of the winning candidate rescans its LDS slice
        int w = bcast;
        if (idx == w) {
            v = -INFINITY; idx = 0x7fffffff;
            for (int q = tid; q < Q_; q += 1024) {
                float s = smem[q];
                if (better(s, q, v, idx)) { v = s; idx = q; }
            }
        }
    }
    __syncthreads();

    // ---- Gather: boxes (K x 4) and sigmoid(logits) (K x 80) ----
    for (int i = tid; i < K_ * C_; i += 1024) {
        int k = i / C_;
        int c = i - k * C_;
        int q = topIdx[k];
        float x = logits[((size_t)b * Q_ + q) * C_ + c];
        out_scores[((size_t)b * K_ + k) * C_ + c] = 1.0f / (1.0f + expf(-x));
    }
    for (int i = tid; i < K_ * 4; i += 1024) {
        int k = i >> 2;
        int c = i & 3;
        int q = topIdx[k];
        out_boxes[((size_t)b * K_ + k) * 4 + c] = boxes[((size_t)b * Q_ + q) * 4 + c];
    }
}

// ---------------------------------------------------------------------------
extern "C" void kernel_launch(void* const* d_in, const int* in_sizes, int n_in,
                              void* d_out, int out_size, void* d_ws, size_t ws_size,
                              hipStream_t stream) {
    (void)in_sizes; (void)n_in; (void)out_size; (void)ws_size;

    const float* boxes  = (const float*)d_in[0];   // (16, 20000, 4)
    const float* logits = (const float*)d_in[1];   // (16, 20000, 80)

    float* out_boxes  = (float*)d_out;                       // (16, 100, 4)
    float* out_scores = (float*)d_out + (size_t)B_ * K_ * 4; // (16, 100, 80)

    float* scores_ws = (float*)d_ws;               // 16 * 20000 floats = 1.28 MB

    int total = B_ * Q_;
    score_kernel<<<(total + 255) / 256, 256, 0, stream>>>(logits, scores_ws, total);

    size_t dynLDS = (size_t)Q_ * sizeof(float);    // 80 KB
    topk_kernel<<<B_, 1024, dynLDS, stream>>>(scores_ws, logits, boxes,
                                              out_boxes, out_scores);
}